// EnsembleModel_61813169324777
// MI455X (gfx1250) — compile-verified
//
#include <hip/hip_runtime.h>

typedef __attribute__((ext_vector_type(2))) float v2f;
typedef __attribute__((ext_vector_type(4))) float v4f;
typedef __attribute__((ext_vector_type(8))) float v8f;

#define ALPHA 0.3f
constexpr int B_  = 16;
constexpr int S_  = 160;
constexpr int R_  = 40;
constexpr int TP_ = 64;     // tag vocab 50 padded to 64
constexpr int KEYS_ = TP_ * TP_;   // 4096

// workspace layout (floats):
//   partial : (R+1) * 4 * KEYS    -- per-(unit, batch-group) partial tables
//   tabArc  : KEYS                -- index t1*64 + t2
//   tabRel  : KEYS * R            -- index (t1*64 + t2)*R + r

// ---------------------------------------------------------------------------
// tab = sum_b M_b^T * A_b * M_b  via V_WMMA_F32_16X16X4_F32 (exact f32 math),
// one-hot operands synthesized in registers from adds[].
// grid = (numUnits, 4); block = 256 (8 waves). blockIdx.y picks 4 batches.
// ESTRIDE: element stride along p2 (1 for a_arc slice, R for an a_rel channel).
// Writes a full 64x64 partial table per workgroup (no atomics -> deterministic).
// ---------------------------------------------------------------------------
template <int ESTRIDE>
__global__ __launch_bounds__(256)
void tab_build_kernel(const float* __restrict__ A0,
                      const int*   __restrict__ adds,
                      float* __restrict__ partial)
{
    __shared__ float sW[S_ * TP_];   // 160*64 f32 = 40 KB
    __shared__ int   sAdds[S_];

    const int u    = blockIdx.x;                // channel (0 for arc kernel)
    const int b0   = blockIdx.y * 4;
    const int lane = threadIdx.x & 31;
    const int wave = threadIdx.x >> 5;          // 0..7

    const int lm    = lane & 15;                // row/col within a 16-wide tile
    const int lkOff = (lane >> 4) * 2;          // K base: 0 (lanes 0-15) / 2 (16-31)
    const int rowAdd = (lane >> 4) * 8;         // C/D row offset for upper lanes

    v8f acc0 = {};                              // two 16x16 tab tiles per wave,
    v8f acc1 = {};                              // accumulated over the 4 batches

    // step-2 tile geometry (fixed per wave): tiles 2w, 2w+1 share t1tile
    const int t1tile = wave >> 1;
    const int t2t0   = (wave * 2) & 3;
    const int t1 = t1tile * 16 + lm;
    const int n0 = t2t0 * 16 + lm;              // tile 2w   t2 column
    const int n1 = n0 + 16;                     // tile 2w+1 t2 column

    for (int bb = 0; bb < 4; ++bb) {
        const int b = b0 + bb;

        for (int i = threadIdx.x; i < S_; i += 256) sAdds[i] = adds[b * S_ + i];
        __syncthreads();

        const float* Ab = A0 + (size_t)b * S_ * S_ * ESTRIDE + u;

        // ---- step 1: W[p1][t2] = sum_p2 A[p1][p2] * (adds[p2]==t2) ----
        // wave w owns 16-row strips mt = w, w+8; A fragment reused over 4 n-tiles
        for (int mt = wave; mt < 10; mt += 8) {
            const int m = mt * 16 + lm;         // p1 row
            v8f c0 = {}, c1 = {}, c2 = {}, c3 = {};
            const float* p = Ab + ((size_t)m * S_ + lkOff) * ESTRIDE;
            for (int kb = 0; kb < S_; kb += 4) {
                const int k = kb + lkOff;       // p2
                v2f a;
                if constexpr (ESTRIDE == 1) {
                    a = *(const v2f*)p;         // global_load_b64
                } else {
                    a.x = p[0];
                    a.y = p[ESTRIDE];
                }
                p += 4 * ESTRIDE;
                const int k0 = sAdds[k], k1 = sAdds[k + 1];
                v2f bm;
                bm.x = (k0 == lm)      ? 1.0f : 0.0f;
                bm.y = (k1 == lm)      ? 1.0f : 0.0f;
                c0 = __builtin_amdgcn_wmma_f32_16x16x4_f32(false, a, false, bm, (short)0, c0, false, false);
                bm.x = (k0 == lm + 16) ? 1.0f : 0.0f;
                bm.y = (k1 == lm + 16) ? 1.0f : 0.0f;
                c1 = __builtin_amdgcn_wmma_f32_16x16x4_f32(false, a, false, bm, (short)0, c1, false, false);
                bm.x = (k0 == lm + 32) ? 1.0f : 0.0f;
                bm.y = (k1 == lm + 32) ? 1.0f : 0.0f;
                c2 = __builtin_amdgcn_wmma_f32_16x16x4_f32(false, a, false, bm, (short)0, c2, false, false);
                bm.x = (k0 == lm + 48) ? 1.0f : 0.0f;
                bm.y = (k1 == lm + 48) ? 1.0f : 0.0f;
                c3 = __builtin_amdgcn_wmma_f32_16x16x4_f32(false, a, false, bm, (short)0, c3, false, false);
            }
#pragma unroll
            for (int r2 = 0; r2 < 8; ++r2) {
                const int row = mt * 16 + r2 + rowAdd;
                sW[row * TP_ + lm]      = c0[r2];
                sW[row * TP_ + lm + 16] = c1[r2];
                sW[row * TP_ + lm + 32] = c2[r2];
                sW[row * TP_ + lm + 48] = c3[r2];
            }
        }
        __syncthreads();

        // ---- step 2: tab[t1][t2] += sum_p1 (adds[p1]==t1) * W[p1][t2] ----
        // one-hot A fragment shared by the wave's two tiles
        for (int kb = 0; kb < S_; kb += 4) {
            const int k = kb + lkOff;           // p1
            v2f a;
            a.x = (sAdds[k]     == t1) ? 1.0f : 0.0f;
            a.y = (sAdds[k + 1] == t1) ? 1.0f : 0.0f;
            v2f bw;
            bw.x = sW[(k)     * TP_ + n0];
            bw.y = sW[(k + 1) * TP_ + n0];
            acc0 = __builtin_amdgcn_wmma_f32_16x16x4_f32(false, a, false, bw, (short)0, acc0, false, false);
            bw.x = sW[(k)     * TP_ + n1];
            bw.y = sW[(k + 1) * TP_ + n1];
            acc1 = __builtin_amdgcn_wmma_f32_16x16x4_f32(false, a, false, bw, (short)0, acc1, false, false);
        }
        __syncthreads();                        // sW reused next batch
    }

    // ---- flush: plain stores into this workgroup's private partial table ----
    float* pout = partial + ((size_t)(u * 4 + blockIdx.y)) * KEYS_;
#pragma unroll
    for (int r2 = 0; r2 < 8; ++r2) {
        const int row = t1tile * 16 + r2 + rowAdd;
        pout[row * TP_ + n0] = acc0[r2];
        pout[row * TP_ + n1] = acc1[r2];
    }
}

// fixed-order reduction of the 4 batch-group partials -> deterministic output
__global__ void tab_reduce_kernel(const float* __restrict__ partial,
                                  float* __restrict__ tabArc,
                                  float* __restrict__ tabRel)
{
    int idx = blockIdx.x * blockDim.x + threadIdx.x;
    if (idx >= (R_ + 1) * KEYS_) return;
    int u   = idx / KEYS_;
    int key = idx % KEYS_;
    const float* p = partial + (size_t)u * 4 * KEYS_ + key;
    float s = ((p[0] + p[KEYS_]) + (p[2 * KEYS_] + p[3 * KEYS_]));
    if (u == R_) tabArc[key] = s;
    else         tabRel[(size_t)key * R_ + u] = s;
}

__global__ void gather_arc_kernel(const float* __restrict__ s_arc,
                                  const int*   __restrict__ pos,
                                  const float* __restrict__ tabArc,
                                  float* __restrict__ out)
{
    int idx = blockIdx.x * blockDim.x + threadIdx.x;
    if (idx >= B_ * S_ * S_) return;
    int j = idx % S_;
    int i = (idx / S_) % S_;
    int b = idx / (S_ * S_);
    int t1 = pos[b * S_ + i];
    int t2 = pos[b * S_ + j];
    float sv = __builtin_nontemporal_load(s_arc + idx);
    __builtin_nontemporal_store(sv + ALPHA * tabArc[t1 * TP_ + t2], out + idx);
}

// one thread per (b,i,j,r-quad): NT float4 streams; tab r-axis contiguous & cached
__global__ void gather_rel_kernel(const float* __restrict__ s_rel,
                                  const int*   __restrict__ pos,
                                  const float* __restrict__ tabRel,
                                  float* __restrict__ out)
{
    constexpr int RQ = R_ / 4;
    int idx = blockIdx.x * blockDim.x + threadIdx.x;
    if (idx >= B_ * S_ * S_ * RQ) return;
    int rq   = idx % RQ;
    int rest = idx / RQ;
    int j = rest % S_;
    int i = (rest / S_) % S_;
    int b = rest / (S_ * S_);
    int t1 = pos[b * S_ + i];
    int t2 = pos[b * S_ + j];
    size_t base = (size_t)rest * R_ + rq * 4;
    v4f sv = __builtin_nontemporal_load((const v4f*)(s_rel + base));
    v4f tv = *(const v4f*)(tabRel + (size_t)(t1 * TP_ + t2) * R_ + rq * 4);
    v4f o  = sv + ALPHA * tv;
    __builtin_nontemporal_store(o, (v4f*)(out + base));
}

extern "C" void kernel_launch(void* const* d_in, const int* in_sizes, int n_in,
                              void* d_out, int out_size, void* d_ws, size_t ws_size,
                              hipStream_t stream)
{
    const float* a_arc = (const float*)d_in[0];
    const float* a_rel = (const float*)d_in[1];
    const float* s_arc = (const float*)d_in[2];
    const float* s_rel = (const float*)d_in[3];
    const int*   adds  = (const int*)d_in[4];
    const int*   pos   = (const int*)d_in[5];
    // d_in[6] = n_tags (known = 50, padded to 64)

    float* out     = (float*)d_out;
    float* partial = (float*)d_ws;                       // (R+1)*4*4096 floats
    float* tabArc  = partial + (size_t)(R_ + 1) * 4 * KEYS_;
    float* tabRel  = tabArc + KEYS_;

    // phase 1: one-hot GEMMs (rel channels u=0..39, arc as u=40 via offset)
    tab_build_kernel<R_><<<dim3(R_, 4), 256, 0, stream>>>(a_rel, adds, partial);
    tab_build_kernel<1><<<dim3(1, 4), 256, 0, stream>>>(
        a_arc, adds, partial + (size_t)R_ * 4 * KEYS_);

    const int nTab = (R_ + 1) * KEYS_;
    tab_reduce_kernel<<<(nTab + 255) / 256, 256, 0, stream>>>(partial, tabArc, tabRel);

    // phase 2: streaming gather + axpy
    const int nArc = B_ * S_ * S_;
    gather_arc_kernel<<<(nArc + 255) / 256, 256, 0, stream>>>(s_arc, pos, tabArc, out);

    const int nRelQ = B_ * S_ * S_ * (R_ / 4);
    gather_rel_kernel<<<(nRelQ + 255) / 256, 256, 0, stream>>>(s_rel, pos, tabRel, out + nArc);
}